// AdaptiveBreadthLayer_60163901882850
// MI455X (gfx1250) — compile-verified
//
#include <hip/hip_runtime.h>
#include <hip/hip_bf16.h>
#include <math.h>

typedef float v2f __attribute__((ext_vector_type(2)));
typedef float v8f __attribute__((ext_vector_type(8)));

#define NNODES 50000
#define NEDGES 800000
#define INDIM  256
#define HEADS  4
#define HDIM   64
#define OUTF   256   // HEADS*HDIM
#define NEG_SLOPE 0.2f

// ---------------------------------------------------------------- init
__global__ void init_ws(float* __restrict__ outacc, float* __restrict__ maxv,
                        float* __restrict__ denom) {
    int i = blockIdx.x * blockDim.x + threadIdx.x;
    if (i < NNODES * OUTF)  outacc[i] = 0.0f;
    if (i < NNODES * HEADS) { maxv[i] = -INFINITY; denom[i] = 0.0f; }
}

// ---------------------------------------------------------------- GEMM: feat = h @ W  (fp32 WMMA)
// One wave computes one 16x16 tile of feat. K stepped by 4 with v_wmma_f32_16x16x4_f32.
__global__ __launch_bounds__(256) void gemm_feat(const float* __restrict__ h,
                                                 const float* __restrict__ W,
                                                 float* __restrict__ feat) {
    const int wave = threadIdx.x >> 5;
    const int lane = threadIdx.x & 31;
    const int tile = blockIdx.x * 8 + wave;            // 3125*16 = 50000 tiles
    if (tile >= (NNODES / 16) * (OUTF / 16)) return;   // wave-uniform guard
    const int mBase = (tile >> 4) << 4;                // tile / 16 * 16
    const int nBase = (tile & 15) << 4;
    const int half  = lane >> 4;                       // 0 or 1
    const int l16   = lane & 15;

    // A layout (16x4 f32): lane l -> row M = l&15 ; VGPR j holds K = j + 2*half
    // B layout (4x16 f32): lane l -> col N = l&15 ; VGPR j holds K = j + 2*half
    const float* aPtr = h + (size_t)(mBase + l16) * INDIM + half * 2;
    const float* bPtr = W + (size_t)(half * 2) * OUTF + (nBase + l16);

    v8f acc = {0.f, 0.f, 0.f, 0.f, 0.f, 0.f, 0.f, 0.f};

    #pragma unroll 4
    for (int k = 0; k < INDIM; k += 4) {
        v2f a = *(const v2f*)(aPtr + k);               // contiguous 8B (K=+0,+1)
        v2f b;
        b.x = bPtr[(size_t)k * OUTF];                  // row K=k+2*half
        b.y = bPtr[(size_t)(k + 1) * OUTF];            // row K=k+1+2*half
        acc = __builtin_amdgcn_wmma_f32_16x16x4_f32(
                  false, a, false, b, (short)0, acc, false, false);
    }

    // C/D layout: VGPR i, lanes 0-15 -> M=i, lanes 16-31 -> M=i+8 ; N = lane&15
    float* o = feat + (size_t)(mBase + half * 8) * OUTF + nBase + l16;
    #pragma unroll
    for (int i = 0; i < 8; ++i) o[(size_t)i * OUTF] = acc[i];
}

// ---------------------------------------------------------------- el/er per (node, head)
// One wave per node: 32 lanes x 8 floats = full 256-wide feat row.
__global__ __launch_bounds__(256) void node_logits(const float* __restrict__ feat,
                                                   const float* __restrict__ attn_l,
                                                   const float* __restrict__ attn_r,
                                                   float* __restrict__ el,
                                                   float* __restrict__ er) {
    const int wave = threadIdx.x >> 5;
    const int lane = threadIdx.x & 31;
    const int n = blockIdx.x * 8 + wave;
    if (n >= NNODES) return;

    const float4* f  = (const float4*)(feat + (size_t)n * OUTF + lane * 8);
    const float4* al = (const float4*)(attn_l + lane * 8);
    const float4* ar = (const float4*)(attn_r + lane * 8);
    float4 f0 = f[0],  f1 = f[1];
    float4 a0 = al[0], a1 = al[1];
    float4 r0 = ar[0], r1 = ar[1];

    float sl = f0.x*a0.x + f0.y*a0.y + f0.z*a0.z + f0.w*a0.w
             + f1.x*a1.x + f1.y*a1.y + f1.z*a1.z + f1.w*a1.w;
    float sr = f0.x*r0.x + f0.y*r0.y + f0.z*r0.z + f0.w*r0.w
             + f1.x*r1.x + f1.y*r1.y + f1.z*r1.z + f1.w*r1.w;

    // reduce within 8-lane groups (one head per group of 8 lanes)
    #pragma unroll
    for (int off = 1; off < 8; off <<= 1) {
        sl += __shfl_xor(sl, off);
        sr += __shfl_xor(sr, off);
    }
    if ((lane & 7) == 0) {
        int hd = lane >> 3;
        el[n * HEADS + hd] = sl;
        er[n * HEADS + hd] = sr;
    }
}

// ---------------------------------------------------------------- edge helpers
__device__ __forceinline__ float leaky(float x) { return x > 0.0f ? x : NEG_SLOPE * x; }

__device__ __forceinline__ void atomicMaxF(float* addr, float val) {
    if (val >= 0.0f) atomicMax((int*)addr, __float_as_int(val));
    else             atomicMin((unsigned int*)addr, __float_as_uint(val));
}

// ---------------------------------------------------------------- segment max over dst
__global__ void edge_max(const int* __restrict__ src, const int* __restrict__ dst,
                         const float* __restrict__ el, const float* __restrict__ er,
                         float* __restrict__ maxv) {
    int t = blockIdx.x * blockDim.x + threadIdx.x;
    if (t >= NEDGES * HEADS) return;
    int e = t >> 2, hd = t & 3;
    int s = src[e], d = dst[e];
    float v = leaky(el[s * HEADS + hd] + er[d * HEADS + hd]);
    atomicMaxF(&maxv[d * HEADS + hd], v);
}

// ---------------------------------------------------------------- segment sum of exp
__global__ void edge_denom(const int* __restrict__ src, const int* __restrict__ dst,
                           const float* __restrict__ el, const float* __restrict__ er,
                           const float* __restrict__ maxv, float* __restrict__ denom) {
    int t = blockIdx.x * blockDim.x + threadIdx.x;
    if (t >= NEDGES * HEADS) return;
    int e = t >> 2, hd = t & 3;
    int s = src[e], d = dst[e];
    float v = leaky(el[s * HEADS + hd] + er[d * HEADS + hd]);
    atomicAdd(&denom[d * HEADS + hd], expf(v - maxv[d * HEADS + hd]));
}

// ---------------------------------------------------------------- weighted scatter-add
// One wave per edge: lane covers 8 consecutive feat elements; head = lane/8.
__global__ __launch_bounds__(256) void edge_aggregate(const int* __restrict__ src,
                                                      const int* __restrict__ dst,
                                                      const float* __restrict__ el,
                                                      const float* __restrict__ er,
                                                      const float* __restrict__ maxv,
                                                      const float* __restrict__ denom,
                                                      const float* __restrict__ feat,
                                                      float* __restrict__ outacc) {
    const int wave = threadIdx.x >> 5;
    const int lane = threadIdx.x & 31;
    const int e = blockIdx.x * 8 + wave;
    if (e >= NEDGES) return;
    const int s = src[e], d = dst[e];
    const int hd   = lane >> 3;
    const int nidx = d * HEADS + hd;

    float v     = leaky(el[s * HEADS + hd] + er[nidx]);
    float alpha = expf(v - maxv[nidx]) / fmaxf(denom[nidx], 1e-9f);

    const float4* f = (const float4*)(feat + (size_t)s * OUTF + lane * 8);
    float4 f0 = f[0], f1 = f[1];
    float* o = outacc + (size_t)d * OUTF + lane * 8;
    atomicAdd(o + 0, alpha * f0.x);
    atomicAdd(o + 1, alpha * f0.y);
    atomicAdd(o + 2, alpha * f0.z);
    atomicAdd(o + 3, alpha * f0.w);
    atomicAdd(o + 4, alpha * f1.x);
    atomicAdd(o + 5, alpha * f1.y);
    atomicAdd(o + 6, alpha * f1.z);
    atomicAdd(o + 7, alpha * f1.w);
}

// ---------------------------------------------------------------- bias + tanh + head mean
__global__ void finalize(const float* __restrict__ outacc, const float* __restrict__ bias,
                         float* __restrict__ out) {
    int t = blockIdx.x * blockDim.x + threadIdx.x;
    if (t >= NNODES * HDIM) return;
    int n = t >> 6, dd = t & 63;
    float sum = 0.0f;
    #pragma unroll
    for (int hh = 0; hh < HEADS; ++hh)
        sum += tanhf(outacc[(size_t)n * OUTF + hh * HDIM + dd] + bias[hh * HDIM + dd]);
    out[t] = sum * 0.25f;
}

// ---------------------------------------------------------------- launcher
extern "C" void kernel_launch(void* const* d_in, const int* in_sizes, int n_in,
                              void* d_out, int out_size, void* d_ws, size_t ws_size,
                              hipStream_t stream) {
    const float* h      = (const float*)d_in[0];
    const float* W      = (const float*)d_in[1];
    const float* attn_l = (const float*)d_in[2];
    const float* attn_r = (const float*)d_in[3];
    const float* bias   = (const float*)d_in[4];
    const int*   src    = (const int*)d_in[5];
    const int*   dst    = (const int*)d_in[6];
    float* out = (float*)d_out;

    float* ws     = (float*)d_ws;
    float* feat   = ws;                                   // N*256
    float* outacc = feat   + (size_t)NNODES * OUTF;       // N*256
    float* el     = outacc + (size_t)NNODES * OUTF;       // N*4
    float* er     = el     + (size_t)NNODES * HEADS;      // N*4
    float* maxv   = er     + (size_t)NNODES * HEADS;      // N*4
    float* denom  = maxv   + (size_t)NNODES * HEADS;      // N*4

    init_ws<<<(NNODES * OUTF + 255) / 256, 256, 0, stream>>>(outacc, maxv, denom);

    // 50000 tiles of 16x16, 8 waves (tiles) per 256-thread block
    gemm_feat<<<(NNODES / 16) * (OUTF / 16) / 8, 256, 0, stream>>>(h, W, feat);

    node_logits<<<(NNODES + 7) / 8, 256, 0, stream>>>(feat, attn_l, attn_r, el, er);

    edge_max  <<<(NEDGES * HEADS + 255) / 256, 256, 0, stream>>>(src, dst, el, er, maxv);
    edge_denom<<<(NEDGES * HEADS + 255) / 256, 256, 0, stream>>>(src, dst, el, er, maxv, denom);

    edge_aggregate<<<(NEDGES + 7) / 8, 256, 0, stream>>>(src, dst, el, er, maxv, denom,
                                                         feat, outacc);

    finalize<<<(NNODES * HDIM + 255) / 256, 256, 0, stream>>>(outacc, bias, out);
}